// BillehColumn_89670327206508
// MI455X (gfx1250) — compile-verified
//
#include <hip/hip_runtime.h>
#include <stdint.h>

// ---------------------------------------------------------------------------
// GLIF neuron step (BillehColumn) for MI455X / gfx1250.
//
// Phase 1: edge SpMM  i_rec[b,row] += w * z[b,col]   (E=5M, B=4)
//   - HBM-bound: 60MB edge stream @ 23.3TB/s ~ 2.6us floor.
//   - Edge stream staged into LDS via the Tensor Data Mover
//     (tensor_load_to_lds, TENSORcnt), double-buffered so DMA of tile t+1
//     overlaps gather+atomic work on tile t.
//   - z gathers and f32 atomic scatters hit L2 (z_buf 4MB, i_rec 3.2MB
//     both resident in the 192MB L2). Atomics skipped when z==0 (~5% spike
//     density -> ~20x fewer atomics).
// Phase 2: fused elementwise neuron update, one thread per (b,n), float4
//   loads/stores on the R=4 synapse axis, writes the concatenated output.
// ---------------------------------------------------------------------------

#define BB 4
#define RR 4
#define DD 5
#define DT_C 1.0f

#define TILE 2048
#define EDGE_BLOCKS 512
#define EDGE_THREADS 256

#if defined(__has_builtin)
#if __has_builtin(__builtin_amdgcn_tensor_load_to_lds) && __has_builtin(__builtin_amdgcn_s_wait_tensorcnt)
#define HAVE_TDM 1
#endif
#endif

#ifdef HAVE_TDM
typedef unsigned int v4u_t __attribute__((ext_vector_type(4)));
typedef int v8i_t __attribute__((ext_vector_type(8)));
typedef int v4i_t __attribute__((ext_vector_type(4)));

// 1-D DMA: copy tile_elems 4-byte elements from gsrc into LDS at lds_off.
// tensor_dim0 = n_valid so the OOB tail reads return zero (zero-padded LDS).
// D# bitfields per CDNA5 ISA ch.8 (group0 128b, group1 256b; groups 2/3 = 0,
// i.e. <=2D tensor). This toolchain exposes the 6-arg builtin form.
__device__ __forceinline__ void tdm_load_1d(const void* gsrc, unsigned lds_off,
                                            unsigned n_valid, unsigned tile_elems) {
  unsigned long long ga = (unsigned long long)gsrc;
  v4u_t g0;
  g0.x = 1u;                                                  // count=1, user desc
  g0.y = lds_off;                                             // lds_addr [63:32]
  g0.z = (unsigned)(ga & 0xFFFFFFFFull);                      // global_addr[31:0]
  g0.w = (unsigned)((ga >> 32) & 0x01FFFFFFull) | (2u << 30); // ga[56:32] | type=2
  v8i_t g1;
  g1[0] = (int)(2u << 16);                          // wg_mask=0, data_size=2 (4B)
  g1[1] = (int)((n_valid & 0xFFFFu) << 16);         // tensor_dim0[15:0]
  g1[2] = (int)((n_valid >> 16) | (1u << 16));      // tensor_dim0[31:16] | td1 lo=1
  g1[3] = (int)(tile_elems << 16);                  // td1 hi=0 | tile_dim0
  g1[4] = 1;                                        // tile_dim1=1, tile_dim2=0
  g1[5] = (int)tile_elems;                          // tensor_dim0_stride lo
  g1[6] = 0;                                        // stride0 hi | stride1 lo
  g1[7] = 0;                                        // stride1 hi
  v4i_t zz4 = {0, 0, 0, 0};
  v8i_t zz8 = {0, 0, 0, 0, 0, 0, 0, 0};
  __builtin_amdgcn_tensor_load_to_lds(g0, g1, zz4, zz4, zz8, 0);
}
#endif

__global__ void __launch_bounds__(EDGE_THREADS)
edge_spmm_kernel(const float* __restrict__ rec_w,
                 const int* __restrict__ rec_rows,
                 const int* __restrict__ rec_cols,
                 const float* __restrict__ z_buf,  // (B, N*D)
                 float* __restrict__ irec,         // (B, N*R) scratch, pre-zeroed
                 int E, int NR, int ND,
                 int per) {                        // edges per workgroup (host-computed)
  __shared__ float s_w[2][TILE];
  __shared__ int s_r[2][TILE];
  __shared__ int s_c[2][TILE];

  long long e0 = (long long)blockIdx.x * (long long)per;
  long long e1 = e0 + (long long)per;
  if (e1 > (long long)E) e1 = (long long)E;
  if (e0 >= e1) return;  // workgroup-uniform: no barrier imbalance
  int nTiles = (int)((e1 - e0 + TILE - 1) / TILE);

#ifdef HAVE_TDM
  bool issuer = (threadIdx.x < 32);  // wave 0 owns the DMA (TDM ignores EXEC)
  if (issuer) {
    long long rem = e1 - e0;
    unsigned nv0 = (unsigned)(rem < (long long)TILE ? rem : (long long)TILE);
    tdm_load_1d(rec_w + e0, (unsigned)(unsigned long long)(void*)&s_w[0][0], nv0, TILE);
    tdm_load_1d(rec_rows + e0, (unsigned)(unsigned long long)(void*)&s_r[0][0], nv0, TILE);
    tdm_load_1d(rec_cols + e0, (unsigned)(unsigned long long)(void*)&s_c[0][0], nv0, TILE);
  }
#endif

  for (int t = 0; t < nTiles; ++t) {
    long long base = e0 + (long long)t * TILE;
    long long remc = e1 - base;
    int nValid = (int)(remc < (long long)TILE ? remc : (long long)TILE);
    int cur = t & 1;

#ifdef HAVE_TDM
    if (issuer) __builtin_amdgcn_s_wait_tensorcnt(0);  // tile t landed in LDS
    __syncthreads();  // all waves: tile t visible; tile t-1 fully consumed
    if (issuer && (t + 1) < nTiles) {                  // overlap DMA t+1 with compute t
      long long nb = base + TILE;
      long long remn = e1 - nb;
      unsigned nv = (unsigned)(remn < (long long)TILE ? remn : (long long)TILE);
      int nxt = (t + 1) & 1;
      tdm_load_1d(rec_w + nb, (unsigned)(unsigned long long)(void*)&s_w[nxt][0], nv, TILE);
      tdm_load_1d(rec_rows + nb, (unsigned)(unsigned long long)(void*)&s_r[nxt][0], nv, TILE);
      tdm_load_1d(rec_cols + nb, (unsigned)(unsigned long long)(void*)&s_c[nxt][0], nv, TILE);
    }
#else
    // Fallback: cooperative staged copy (single-buffered into slot `cur`).
    for (int i = threadIdx.x; i < nValid; i += EDGE_THREADS) {
      s_w[cur][i] = rec_w[base + i];
      s_r[cur][i] = rec_rows[base + i];
      s_c[cur][i] = rec_cols[base + i];
    }
    __syncthreads();
#endif

    for (int i = threadIdx.x; i < nValid; i += EDGE_THREADS) {
      float w = s_w[cur][i];
      int row = s_r[cur][i];
      int col = s_c[cur][i];
      const float* zp = z_buf + col;
      float z0 = zp[0];
      float z1 = zp[ND];
      float z2 = zp[2 * ND];
      float z3 = zp[3 * ND];
      // spikes are ~5% dense: skip zero contributions
      if (z0 != 0.0f) atomicAdd(irec + row, w * z0);
      if (z1 != 0.0f) atomicAdd(irec + NR + row, w * z1);
      if (z2 != 0.0f) atomicAdd(irec + 2 * NR + row, w * z2);
      if (z3 != 0.0f) atomicAdd(irec + 3 * NR + row, w * z3);
    }
#ifndef HAVE_TDM
    __syncthreads();  // buffer reuse guard for fallback path
#endif
  }
}

__global__ void __launch_bounds__(256)
neuron_update_kernel(const float* __restrict__ inputs,      // (B, N*R)
                     const float* __restrict__ z_buf,       // (B, N*D)
                     const float* __restrict__ v_in,        // (B, N)
                     const float* __restrict__ r_in,        // (B, N)
                     const float* __restrict__ asc1_in,     // (B, N)
                     const float* __restrict__ asc2_in,     // (B, N)
                     const float* __restrict__ psc_rise_in, // (B, N*R)
                     const float* __restrict__ psc_in,      // (B, N*R)
                     const float* __restrict__ irec,        // (B, N*R)
                     const float* __restrict__ syn_decay,   // (N, R)
                     const float* __restrict__ psc_initial, // (N, R)
                     const float* __restrict__ t_ref,       // (N)
                     const float* __restrict__ asc_amps,    // (N, 2)
                     const float* __restrict__ kparam,      // (N, 2)
                     const float* __restrict__ v_th,
                     const float* __restrict__ e_l,
                     const float* __restrict__ v_reset,
                     const float* __restrict__ g_p,
                     const float* __restrict__ decay_p,
                     const float* __restrict__ current_factor,
                     const float* __restrict__ voltage_scale,
                     const float* __restrict__ voltage_offset,
                     float* __restrict__ out, int N) {
  int gid = blockIdx.x * blockDim.x + threadIdx.x;
  if (gid >= BB * N) return;
  int b = gid / N;
  int n = gid - b * N;
  int NR = N * RR;
  int ND = N * DD;
  long long bN = (long long)b * N;
  long long bNR = (long long)b * NR;
  long long bND = (long long)b * ND;

  // R=4 synapse axis: 128-bit loads
  float4 pr = *(const float4*)(psc_rise_in + bNR + (long long)n * 4);
  float4 pc = *(const float4*)(psc_in + bNR + (long long)n * 4);
  float4 in4 = *(const float4*)(inputs + bNR + (long long)n * 4);
  float4 ir4 = *(const float4*)(irec + bNR + (long long)n * 4);
  float4 sd = *(const float4*)(syn_decay + (long long)n * 4);
  float4 pi = *(const float4*)(psc_initial + (long long)n * 4);

  float4 npr, npc;
  npr.x = sd.x * pr.x + (in4.x + ir4.x) * pi.x;
  npr.y = sd.y * pr.y + (in4.y + ir4.y) * pi.y;
  npr.z = sd.z * pr.z + (in4.z + ir4.z) * pi.z;
  npr.w = sd.w * pr.w + (in4.w + ir4.w) * pi.w;
  npc.x = pc.x * sd.x + DT_C * sd.x * pr.x;
  npc.y = pc.y * sd.y + DT_C * sd.y * pr.y;
  npc.z = pc.z * sd.z + DT_C * sd.z * pr.z;
  npc.w = pc.w * sd.w + DT_C * sd.w * pr.w;

  float prev_z = z_buf[bND + n];  // delay slot 0
  float r_old = r_in[bN + n];
  float new_r = fmaxf(r_old + prev_z * t_ref[n] - DT_C, 0.0f);

  float2 kk = *(const float2*)(kparam + 2LL * n);
  float2 amps = *(const float2*)(asc_amps + 2LL * n);
  float sig0 = 1.0f / (1.0f + expf(-kk.x));
  float sig1 = 1.0f / (1.0f + expf(-kk.y));
  float a1 = asc1_in[bN + n];
  float a2 = asc2_in[bN + n];
  float na1 = expf(-DT_C * sig0) * a1 + prev_z * amps.x;
  float na2 = expf(-DT_C * sig1) * a2 + prev_z * amps.y;

  float vth = v_th[n], el = e_l[n];
  float reset_cur = prev_z * (v_reset[n] - vth);
  float input_cur = pc.x + pc.y + pc.z + pc.w;  // OLD psc sum
  float c1 = input_cur + a1 + a2 + g_p[n] * el;
  float nv = decay_p[n] * v_in[bN + n] + current_factor[n] * c1 + reset_cur;
  float v_sc = (nv - vth) / (vth - el);
  float nz = (v_sc > 0.0f) ? 1.0f : 0.0f;
  if (new_r > 0.0f) nz = 0.0f;
  float ov = nv * voltage_scale[n] + voltage_offset[n];

  // Output: concat along last axis per batch: row stride = (5 + 2R + D)*N
  long long ob = (long long)b * (long long)N * (5 + 2 * RR + DD);
  out[ob + n] = nz;
  out[ob + N + n] = ov;
  out[ob + 2LL * N + n] = new_r;
  out[ob + 3LL * N + n] = na1;
  out[ob + 4LL * N + n] = na2;
  *(float4*)(out + ob + 5LL * N + (long long)n * 4) = npr;
  *(float4*)(out + ob + 5LL * N + NR + (long long)n * 4) = npc;
  long long zb = ob + 5LL * N + 2LL * NR;
  out[zb + n] = nz;  // new z at delay 0
#pragma unroll
  for (int d = 1; d < DD; ++d)
    out[zb + (long long)d * N + n] = z_buf[bND + (long long)(d - 1) * N + n];
}

extern "C" void kernel_launch(void* const* d_in, const int* in_sizes, int n_in,
                              void* d_out, int out_size, void* d_ws, size_t ws_size,
                              hipStream_t stream) {
  const float* inputs = (const float*)d_in[0];
  const float* z_buf = (const float*)d_in[1];
  const float* v = (const float*)d_in[2];
  const float* r = (const float*)d_in[3];
  const float* asc1 = (const float*)d_in[4];
  const float* asc2 = (const float*)d_in[5];
  const float* psc_rise = (const float*)d_in[6];
  const float* psc = (const float*)d_in[7];
  const float* rec_w = (const float*)d_in[8];
  const int* rec_rows = (const int*)d_in[9];
  const int* rec_cols = (const int*)d_in[10];
  const float* syn_decay = (const float*)d_in[11];
  const float* psc_initial = (const float*)d_in[12];
  const float* t_ref = (const float*)d_in[13];
  const float* asc_amps = (const float*)d_in[14];
  const float* kparam = (const float*)d_in[15];
  const float* v_th = (const float*)d_in[16];
  const float* e_l = (const float*)d_in[17];
  const float* v_reset = (const float*)d_in[18];
  const float* g = (const float*)d_in[19];
  const float* decay = (const float*)d_in[20];
  const float* current_factor = (const float*)d_in[21];
  const float* voltage_scale = (const float*)d_in[22];
  const float* voltage_offset = (const float*)d_in[23];

  int N = in_sizes[13];  // t_ref has N elements
  int E = in_sizes[8];   // rec_w has E elements
  int NR = N * RR;
  int ND = N * DD;

  float* irec = (float*)d_ws;  // (B, N*R) accumulator
  (void)hipMemsetAsync(irec, 0, (size_t)BB * (size_t)NR * sizeof(float), stream);

  int nTilesTotal = (E + TILE - 1) / TILE;
  int blocks = nTilesTotal < EDGE_BLOCKS ? nTilesTotal : EDGE_BLOCKS;
  int per = (E + blocks - 1) / blocks;  // edges per workgroup, host-side div
  edge_spmm_kernel<<<blocks, EDGE_THREADS, 0, stream>>>(rec_w, rec_rows, rec_cols,
                                                        z_buf, irec, E, NR, ND, per);

  int total = BB * N;
  neuron_update_kernel<<<(total + 255) / 256, 256, 0, stream>>>(
      inputs, z_buf, v, r, asc1, asc2, psc_rise, psc, irec, syn_decay,
      psc_initial, t_ref, asc_amps, kparam, v_th, e_l, v_reset, g, decay,
      current_factor, voltage_scale, voltage_offset, (float*)d_out, N);
}